// GE2ELoss_40046275067922
// MI455X (gfx1250) — compile-verified
//
#include <hip/hip_runtime.h>
#include <hip/hip_bf16.h>
#include <math.h>

typedef __attribute__((ext_vector_type(2))) float v2f;
typedef __attribute__((ext_vector_type(8))) float v8f;

#define NN 2048
#define MM 10
#define DD 256
#define RR (NN * MM)      // 20480 rows of the similarity matrix
#define ASTRIDE 260       // LDS row stride (floats), padded for bank spread

// ---------------------------------------------------------------------------
// Kernel 0: zero the scalar output (harness poisons d_out with 0xAA).
// ---------------------------------------------------------------------------
__global__ void ge2e_zero(float* out) {
    if (blockIdx.x == 0 && threadIdx.x == 0) out[0] = 0.0f;
}

// ---------------------------------------------------------------------------
// Block-wide sum over 256 threads (8 wave32). Internal barriers included.
// ---------------------------------------------------------------------------
__device__ __forceinline__ float blk_sum(float v, float* lds8) {
    __syncthreads();  // protect lds8 from previous use
    #pragma unroll
    for (int off = 16; off >= 1; off >>= 1)
        v += __shfl_xor(v, off, 32);
    if ((threadIdx.x & 31) == 0) lds8[threadIdx.x >> 5] = v;
    __syncthreads();
    if (threadIdx.x == 0) {
        float s = 0.0f;
        for (int i = 0; i < 8; ++i) s += lds8[i];
        lds8[0] = s;
    }
    __syncthreads();
    return lds8[0];
}

// ---------------------------------------------------------------------------
// Kernel 1: per-speaker centroid (mean over M, l2norm) + l2norm of each
// embedding row. One block per speaker n; thread = dimension d (DD == 256).
//   Cn : (NN, DD)  normalized centroids
//   En : (RR, DD)  normalized embeddings
// ---------------------------------------------------------------------------
__global__ __launch_bounds__(256) void ge2e_normalize(
    const float* __restrict__ emb, float* __restrict__ Cn,
    float* __restrict__ En) {
    __shared__ float lds8[8];
    const int n = blockIdx.x;
    const int d = threadIdx.x;

    float e[MM];
    float mean = 0.0f;
    const float* base = emb + ((size_t)n * MM) * DD + d;
    #pragma unroll
    for (int m = 0; m < MM; ++m) {
        e[m] = base[(size_t)m * DD];
        mean += e[m];
    }
    mean *= (1.0f / (float)MM);

    float ssc  = blk_sum(mean * mean, lds8);
    float invc = 1.0f / fmaxf(sqrtf(ssc), 1e-12f);
    Cn[(size_t)n * DD + d] = mean * invc;

    #pragma unroll 1
    for (int m = 0; m < MM; ++m) {
        float ss  = blk_sum(e[m] * e[m], lds8);
        float inv = 1.0f / fmaxf(sqrtf(ss), 1e-12f);
        En[((size_t)n * MM + m) * DD + d] = e[m] * inv;
    }
}

// ---------------------------------------------------------------------------
// Kernel 2: fused GEMM (f32 WMMA 16x16x4) + shifted logsumexp + diag gather.
// Block = 8 waves, owns rows [rb, rb+16) of Enorm (A tile staged in LDS; the
// compiler hoists the 64 A-operand slices into registers). Each wave works
// TWO 16-column tiles per iteration (two independent WMMA accumulator
// chains to hide matrix-pipe latency): cols = wave*32 + iter*256 + {0,16}.
// Fixed logsumexp shift s = |w|+b (dot in [-1,1] => sim <= s): no online max.
// ---------------------------------------------------------------------------
__global__ __launch_bounds__(256) void ge2e_gemm_lse(
    const float* __restrict__ En, const float* __restrict__ Cn,
    const float* __restrict__ wp, const float* __restrict__ bparm,
    float* __restrict__ out) {
    __shared__ float Atile[16 * ASTRIDE];
    __shared__ float red_sum[16];
    __shared__ float red_pos[16];

    const int tid  = threadIdx.x;
    const int lane = tid & 31;
    const int wave = tid >> 5;
    const int rb   = blockIdx.x * 16;

    if (tid < 16) { red_sum[tid] = 0.0f; red_pos[tid] = 0.0f; }

    // Cooperative load of the 16x256 A tile into padded LDS.
    {
        const int r = tid >> 4;          // 0..15 row
        const int c = (tid & 15) * 16;   // 0..240 column chunk
        const float* src = En + ((size_t)(rb + r)) * DD + c;
        float* dst = Atile + r * ASTRIDE + c;
        #pragma unroll
        for (int q = 0; q < 16; ++q) dst[q] = src[q];
    }
    __syncthreads();

    const float aw = fabsf(wp[0]);
    const float bb = bparm[0];
    const float s  = aw + bb;            // fixed logsumexp shift (sim <= s)

    const int   mrow = lane & 15;        // M index for A operand
    const int   kb   = (lane >> 4) * 2;  // K sub-offset per lane half
    const float* ap  = Atile + mrow * ASTRIDE + kb;
    const int rowbase = rb + 8 * (lane >> 4);  // C/D row base for this half

    // Per-row diagonal column (constant across the whole loop).
    int diagc[8];
    #pragma unroll
    for (int v = 0; v < 8; ++v) diagc[v] = (rowbase + v) / MM;

    float sume[8], pose[8];
    #pragma unroll
    for (int v = 0; v < 8; ++v) { sume[v] = 0.0f; pose[v] = 0.0f; }

    #pragma unroll 1
    for (int iter = 0; iter < 8; ++iter) {
        const int col0  = wave * 32 + iter * 256;
        const int ncola = col0 + mrow;        // tile-0 column this lane feeds
        const int ncolb = ncola + 16;         // tile-1 column this lane feeds
        const float* bpa = Cn + (size_t)ncola * DD + kb;
        const float* bpb = Cn + (size_t)ncolb * DD + kb;

        v8f acc0 = {};
        v8f acc1 = {};
        #pragma unroll
        for (int t = 0; t < 64; ++t) {        // K = 256 in steps of 4
            v2f a  = *(const v2f*)(ap  + 4 * t);  // hoisted to VGPRs by LLVM
            v2f b0 = *(const v2f*)(bpa + 4 * t);  // global_load_b64 (L2-hot)
            v2f b1 = *(const v2f*)(bpb + 4 * t);
            acc0 = __builtin_amdgcn_wmma_f32_16x16x4_f32(
                false, a, false, b0, (short)0, acc0, false, false);
            acc1 = __builtin_amdgcn_wmma_f32_16x16x4_f32(
                false, a, false, b1, (short)0, acc1, false, false);
        }

        #pragma unroll
        for (int v = 0; v < 8; ++v) {
            float xa = fmaf(aw, acc0[v], bb);     // sim = |w|*dot + b
            float xb = fmaf(aw, acc1[v], bb);
            sume[v] += __expf(xa - s) + __expf(xb - s);
            pose[v] += (ncola == diagc[v]) ? xa : 0.0f;
            pose[v] += (ncolb == diagc[v]) ? xb : 0.0f;
        }
    }

    // Reduce within each 16-lane group (two independent row halves).
    #pragma unroll
    for (int v = 0; v < 8; ++v) {
        float sv = sume[v], pv = pose[v];
        #pragma unroll
        for (int off = 8; off >= 1; off >>= 1) {
            sv += __shfl_xor(sv, off, 16);
            pv += __shfl_xor(pv, off, 16);
        }
        if ((lane & 15) == 0) {
            const int row16 = v + 8 * (lane >> 4);
            atomicAdd(&red_sum[row16], sv);   // ds_add_f32 across 8 waves
            atomicAdd(&red_pos[row16], pv);
        }
    }
    __syncthreads();

    if (tid < 16) {
        const float loss = __logf(red_sum[tid]) + s - red_pos[tid];
        atomicAdd(out, loss * (1.0f / (float)RR));  // global_atomic_add_f32
    }
}

// ---------------------------------------------------------------------------
extern "C" void kernel_launch(void* const* d_in, const int* in_sizes, int n_in,
                              void* d_out, int out_size, void* d_ws,
                              size_t ws_size, hipStream_t stream) {
    (void)in_sizes; (void)n_in; (void)out_size; (void)ws_size;
    const float* emb = (const float*)d_in[0];  // (N, M, D) fp32
    const float* w   = (const float*)d_in[1];  // scalar
    const float* b   = (const float*)d_in[2];  // scalar
    float* out = (float*)d_out;                // scalar loss

    float* Cn = (float*)d_ws;                  // 2048*256 f32  (2 MB)
    float* En = Cn + (size_t)NN * DD;          // 20480*256 f32 (21 MB)

    ge2e_zero<<<1, 1, 0, stream>>>(out);
    ge2e_normalize<<<NN, 256, 0, stream>>>(emb, Cn, En);
    ge2e_gemm_lse<<<RR / 16, 256, 0, stream>>>(En, Cn, w, b, out);
}